// TypeAreaScaledDotProductAttention_Big_84628035600748
// MI455X (gfx1250) — compile-verified
//
#include <hip/hip_runtime.h>

// ---------------------------------------------------------------------------
// Causal attention, B=8 H=8 L=1024 D=64, q/k/v are sums of 7 tensors.
// Outputs: output [B,H,L,D] then attn [B,H,L,L] concatenated in d_out.
// Memory-bound (≈0.67 GB total traffic, dominated by the 256MB attn write),
// so we keep FP32 precision and use V_WMMA_F32_16X16X4_F32 for both matmuls.
// ---------------------------------------------------------------------------

typedef __attribute__((ext_vector_type(2))) float v2f;
typedef __attribute__((ext_vector_type(4))) float v4f;
typedef __attribute__((ext_vector_type(8))) float v8f;

#define B_ 8
#define H_ 8
#define L_ 1024
#define D_ 64
#define NTOT (B_ * H_ * L_ * D_)   // 4,194,304 elements per tensor

#define INV_T   0.125f             // 1/sqrt(64)
#define NEG_BIG (-1e9f)

// --------------------------- 7-way sum (float4) ----------------------------
__global__ __launch_bounds__(256)
void sum7_kernel(const v4f* __restrict__ a0, const v4f* __restrict__ a1,
                 const v4f* __restrict__ a2, const v4f* __restrict__ a3,
                 const v4f* __restrict__ a4, const v4f* __restrict__ a5,
                 const v4f* __restrict__ a6, v4f* __restrict__ out, int n4) {
  int i = blockIdx.x * blockDim.x + threadIdx.x;
  if (i >= n4) return;
  v4f s = a0[i] + a1[i] + a2[i] + a3[i] + a4[i] + a5[i] + a6[i];
  out[i] = s;
}

// ------------------------ fused causal attention ---------------------------
// Grid: B*H*8 blocks of 256 threads (8 waves). Each wave owns one 16-row
// q-tile and loops over causal key tiles. Two passes:
//   pass 1: S = QK^T/8 (16x v_wmma_f32_16x16x4_f32 per tile) + online
//           row max / sum-of-exp via shfl_xor reductions (wave32 halves).
//   pass 2: recompute S, P = exp(S - m)/sum, write attn, stage P in a
//           padded LDS strip (stride 18 -> bank-conflict-free, b64-aligned),
//           reload as A-fragments, O += P @ V via WMMA.
__global__ __launch_bounds__(256)
void attn_kernel(const float* __restrict__ Q, const float* __restrict__ K,
                 const float* __restrict__ V, float* __restrict__ Out,
                 float* __restrict__ Attn) {
  __shared__ float lds[8 * 16 * 18];     // one 16x18 strip per wave (9 KB)

  const int lane = threadIdx.x & 31;
  const int wave = threadIdx.x >> 5;
  const int half = lane >> 4;            // 0: lanes 0-15, 1: lanes 16-31
  const int lr   = lane & 15;

  const int bh    = blockIdx.x >> 3;                 // 0..63
  const int qtile = ((blockIdx.x & 7) << 3) + wave;  // 0..63
  const int q0    = qtile << 4;

  const float* __restrict__ Qb = Q + (size_t)bh * (L_ * D_);
  const float* __restrict__ Kb = K + (size_t)bh * (L_ * D_);
  const float* __restrict__ Vb = V + (size_t)bh * (L_ * D_);
  float* __restrict__ Ob = Out  + (size_t)bh * (L_ * D_);
  float* __restrict__ Ab = Attn + (size_t)bh * (L_ * L_);
  float* __restrict__ myLds = lds + wave * (16 * 18);

  // ---- load Q tile as A-fragments: lane l holds row (l&15), dims
  //      4*kc + 2*half + {0,1}  (matches 32-bit A 16x4 layout) ----
  v2f aq[16];
  {
    const float* qrow = Qb + (size_t)(q0 + lr) * D_ + (half << 1);
#pragma unroll
    for (int kc = 0; kc < 16; ++kc)
      aq[kc] = *(const v2f*)(qrow + (kc << 2));
  }

  const int jmax = qtile;  // inclusive causal bound on key tiles

  // ------------------------------ pass 1 ----------------------------------
  float rmax[8], rsum[8];
#pragma unroll
  for (int v = 0; v < 8; ++v) { rmax[v] = -__builtin_inff(); rsum[v] = 0.f; }

  for (int j = 0; j <= jmax; ++j) {
    const int k0 = j << 4;
    v8f c = {0.f, 0.f, 0.f, 0.f, 0.f, 0.f, 0.f, 0.f};
    const float* krow = Kb + (size_t)(k0 + lr) * D_ + (half << 1);
#pragma unroll
    for (int kc = 0; kc < 16; ++kc) {
      v2f bk = *(const v2f*)(krow + (kc << 2));  // B = K^T: same pattern as A
      c = __builtin_amdgcn_wmma_f32_16x16x4_f32(false, aq[kc], false, bk,
                                                (short)0, c, false, false);
    }
    const int kk = k0 + lr;                      // global key index (C: N=lane)
#pragma unroll
    for (int v = 0; v < 8; ++v) {
      const int qq = q0 + v + (half << 3);       // C layout: M = v + 8*half
      float s = (kk <= qq) ? c[v] * INV_T : NEG_BIG;
      float m = s;                               // max over 16-lane half
      m = fmaxf(m, __shfl_xor(m, 1, 32));
      m = fmaxf(m, __shfl_xor(m, 2, 32));
      m = fmaxf(m, __shfl_xor(m, 4, 32));
      m = fmaxf(m, __shfl_xor(m, 8, 32));
      const float newm = fmaxf(rmax[v], m);
      float e = __expf(s - newm);
      e += __shfl_xor(e, 1, 32);
      e += __shfl_xor(e, 2, 32);
      e += __shfl_xor(e, 4, 32);
      e += __shfl_xor(e, 8, 32);
      rsum[v] = rsum[v] * __expf(rmax[v] - newm) + e;
      rmax[v] = newm;
    }
  }
  float rinv[8];
#pragma unroll
  for (int v = 0; v < 8; ++v) rinv[v] = 1.0f / rsum[v];

  // ------------------------------ pass 2 ----------------------------------
  v8f co[4];
#pragma unroll
  for (int dt = 0; dt < 4; ++dt) co[dt] = (v8f){0.f,0.f,0.f,0.f,0.f,0.f,0.f,0.f};

  for (int j = 0; j <= jmax; ++j) {
    const int k0 = j << 4;
    v8f c = {0.f, 0.f, 0.f, 0.f, 0.f, 0.f, 0.f, 0.f};
    const float* krow = Kb + (size_t)(k0 + lr) * D_ + (half << 1);
#pragma unroll
    for (int kc = 0; kc < 16; ++kc) {
      v2f bk = *(const v2f*)(krow + (kc << 2));
      c = __builtin_amdgcn_wmma_f32_16x16x4_f32(false, aq[kc], false, bk,
                                                (short)0, c, false, false);
    }
    const int kk = k0 + lr;
#pragma unroll
    for (int v = 0; v < 8; ++v) {
      const int qq = q0 + v + (half << 3);
      float s = (kk <= qq) ? c[v] * INV_T : NEG_BIG;
      float p = __expf(s - rmax[v]) * rinv[v];   // masked -> exactly 0
      Ab[(size_t)qq * L_ + kk] = p;              // normalized attn out
      myLds[(v + (half << 3)) * 18 + lr] = p;    // stage for layout transpose
    }
    // Wave-local LDS RAW fence (DS ops are in-order per wave; this is cheap
    // insurance + a compiler reordering barrier).
    asm volatile("s_wait_dscnt 0" ::: "memory");

    // O += P @ V : A = P from LDS (b64-aligned: even word index), B = V tile.
#pragma unroll
    for (int kc = 0; kc < 4; ++kc) {
      const int kb = (kc << 2) + (half << 1);    // key offset within tile
      v2f ap = *(const v2f*)(myLds + lr * 18 + kb);
      const float* vrow = Vb + (size_t)(k0 + kb) * D_ + lr;
#pragma unroll
      for (int dt = 0; dt < 4; ++dt) {
        v2f bv;
        bv.x = vrow[dt * 16];                    // V[k0+kb  ][dt*16+lr]
        bv.y = vrow[dt * 16 + D_];               // V[k0+kb+1][dt*16+lr]
        co[dt] = __builtin_amdgcn_wmma_f32_16x16x4_f32(false, ap, false, bv,
                                                       (short)0, co[dt],
                                                       false, false);
      }
    }
  }

  // ---- store O (C layout: row v+8*half, col dt*16+lane) ----
#pragma unroll
  for (int dt = 0; dt < 4; ++dt)
#pragma unroll
    for (int v = 0; v < 8; ++v)
      Ob[(size_t)(q0 + v + (half << 3)) * D_ + dt * 16 + lr] = co[dt][v];

  // ---- zero the strictly-upper (non-causal) attn region we own:
  //      rows q0..q0+15, cols q0+16..L-1, wide 16B stores ----
  {
    const v4f z = {0.f, 0.f, 0.f, 0.f};
    float* row = Ab + (size_t)(q0 + lr) * L_;
    for (int col = q0 + 16 + (half << 2); col < L_; col += 8)
      *(v4f*)(row + col) = z;
  }
}

// ---------------------------------------------------------------------------
extern "C" void kernel_launch(void* const* d_in, const int* in_sizes, int n_in,
                              void* d_out, int out_size, void* d_ws, size_t ws_size,
                              hipStream_t stream) {
  (void)in_sizes; (void)n_in; (void)out_size; (void)ws_size;

  // d_ws: q_sum | k_sum | v_sum  (3 * 16 MB = 48 MB)
  float* qsum = (float*)d_ws;
  float* ksum = qsum + NTOT;
  float* vsum = ksum + NTOT;

  const int n4 = NTOT / 4;           // 1,048,576 float4 elements
  dim3 sgrid(n4 / 256), sblk(256);
#define IN4(i) ((const v4f*)d_in[i])
  // q inputs: 0,3,6,9,12,15,18 ; k: 1,4,7,10,13,16,19 ; v: 2,5,8,11,14,17,20
  sum7_kernel<<<sgrid, sblk, 0, stream>>>(IN4(0), IN4(3), IN4(6), IN4(9),
                                          IN4(12), IN4(15), IN4(18),
                                          (v4f*)qsum, n4);
  sum7_kernel<<<sgrid, sblk, 0, stream>>>(IN4(1), IN4(4), IN4(7), IN4(10),
                                          IN4(13), IN4(16), IN4(19),
                                          (v4f*)ksum, n4);
  sum7_kernel<<<sgrid, sblk, 0, stream>>>(IN4(2), IN4(5), IN4(8), IN4(11),
                                          IN4(14), IN4(17), IN4(20),
                                          (v4f*)vsum, n4);
#undef IN4

  float* out  = (float*)d_out;       // [B,H,L,D]
  float* attn = out + NTOT;          // [B,H,L,L]
  attn_kernel<<<dim3(B_ * H_ * 8), dim3(256), 0, stream>>>(qsum, ksum, vsum,
                                                           out, attn);
}